// GNLayer_49022756716605
// MI455X (gfx1250) — compile-verified
//
#include <hip/hip_runtime.h>

typedef __attribute__((ext_vector_type(2))) float v2f;
typedef __attribute__((ext_vector_type(8))) float v8f;
typedef __attribute__((ext_vector_type(4), aligned(8)))  float v4f_g;  // global, 8B-aligned
typedef __attribute__((ext_vector_type(2), aligned(8)))  float v2f_g;  // global, 8B-aligned
typedef __attribute__((ext_vector_type(4), aligned(16))) float v4f_l;  // LDS, 16B-aligned
typedef __attribute__((ext_vector_type(2), aligned(8)))  float v2f_l;  // LDS, 8B-aligned

constexpr int BS   = 1024;
constexpr int NPTS = 4096;
constexpr int DDIM = 3;
constexpr int KDIM = 6;
constexpr int ROWS = NPTS * DDIM;        // 12288 residual rows per batch
constexpr int ACC_STRIDE = 48;           // 6 rows x 8 floats; cols 0..5 = JtJ, col 6 = Jte
constexpr int BLOCKS_PER_BATCH = 4;
constexpr int THREADS = 256;             // 8 wave32 waves
constexpr int WAVES = THREADS / 32;
constexpr float DIAG_MIN = 1e-6f;

// ---------------------------------------------------------------------------
// Kernel 1: zero the per-batch accumulators (d_ws is poisoned by the harness)
// ---------------------------------------------------------------------------
__global__ void gn_zero_acc(float* __restrict__ acc) {
    int i = blockIdx.x * blockDim.x + threadIdx.x;
    if (i < BS * ACC_STRIDE) acc[i] = 0.0f;
}

// ---------------------------------------------------------------------------
// Kernel 2: streaming WMMA accumulation of [JtJ | Jte] per batch.
//
// Chunked pipeline, 32 rows per wave per chunk:
//   stage:   lane l loads row (chunkRow+l): J[row][0..5] (b128+b64), e[row],
//            w[row/3]; stores LDS row = {J0..J5, e, w}  (8 floats, 32 B).
//   compute: 8x V_WMMA_F32_16X16X4_F32, 4 rows each:
//              A[m][k] = weight of row_k times LDS row row_k, col m   (A-lane: K=2*half+{0,1})
//              B[k][n] = LDS row row_k, col n              (B VGPR v: K=v+2*half)
//            -> D[m][n] accumulates  sum w*J[.][m]*J[.][n]  (n<6)
//               D[m][6] accumulates  sum w*J[.][m]*e[.]
//            Lanes m>=7 / cols n>=7 produce garbage confined to unread D
//            entries, so no masking is required.
//   All LDS read offsets are compile-time immediates (pairs 32 B apart merge
//   into ds_load_2addr); globals for chunk+1 are issued before the compute.
// ---------------------------------------------------------------------------
__global__ __launch_bounds__(THREADS)
void gn_accumulate(const float* __restrict__ e,
                   const float* __restrict__ J,
                   const float* __restrict__ weight,
                   float* __restrict__ acc) {
    constexpr int CHUNK  = 32;                 // rows staged per wave per chunk
    constexpr int LROW   = 8;                  // floats per LDS row
    constexpr int LSLICE = 33 * LROW;          // 33 rows: 32 + 1 pad for m<=15 reads

    __shared__ float lds[WAVES][2][LSLICE];    // double-buffered per-wave slices

    const int b    = blockIdx.x / BLOCKS_PER_BATCH;
    const int blk  = blockIdx.x % BLOCKS_PER_BATCH;
    const int wave = threadIdx.x >> 5;
    const int lane = threadIdx.x & 31;
    const int m    = lane & 15;                // A-row / B-col index
    const int half = lane >> 4;                // K-pair select

    const float* __restrict__ Jb = J      + (size_t)b * ROWS * KDIM;
    const float* __restrict__ eb = e      + (size_t)b * ROWS;
    const float* __restrict__ wb = weight + (size_t)b * NPTS;

    const int rowsPerBlock = ROWS / BLOCKS_PER_BATCH;  // 3072
    const int rowsPerWave  = rowsPerBlock / WAVES;     // 384
    const int chunks       = rowsPerWave / CHUNK;      // 12
    const int rowStart     = blk * rowsPerBlock + wave * rowsPerWave;

    float* wbuf = &lds[wave][0][0];
    // Per-lane LDS gather pointers (constant across the whole kernel):
    // element (row_k, m) lives at wbuf[buf*LSLICE + (k*4 + 2*half + j)*LROW + m]
    const float* jcol = wbuf + (2 * half) * LROW + m;   // J/e column m
    const float* wcol = wbuf + (2 * half) * LROW + 7;   // per-row weight

    v8f c0 = {0.f, 0.f, 0.f, 0.f, 0.f, 0.f, 0.f, 0.f};
    v8f c1 = {0.f, 0.f, 0.f, 0.f, 0.f, 0.f, 0.f, 0.f};

    // ---- preload chunk 0 into registers ----
    v4f_g j03; v2f_g j45; float ev, wv;
    {
        const int row = rowStart + lane;
        j03 = *(const v4f_g*)(Jb + (size_t)row * KDIM);       // J[row][0..3]
        j45 = *(const v2f_g*)(Jb + (size_t)row * KDIM + 4);   // J[row][4..5]
        ev  = eb[row];
        wv  = wb[row / DDIM];
    }

    for (int ch = 0; ch < chunks; ++ch) {
        const int buf = ch & 1;

        // ---- stage current chunk into this wave's LDS slice ----
        float* slot = wbuf + buf * LSLICE + lane * LROW;      // 32B-aligned
        *(v4f_l*)(slot)     = v4f_l{j03.x, j03.y, j03.z, j03.w};
        *(v2f_l*)(slot + 4) = v2f_l{j45.x, j45.y};
        *(v2f_l*)(slot + 6) = v2f_l{ev, wv};

        // ---- issue next chunk's global loads (overlap with compute) ----
        if (ch + 1 < chunks) {
            const int row = rowStart + (ch + 1) * CHUNK + lane;
            j03 = *(const v4f_g*)(Jb + (size_t)row * KDIM);
            j45 = *(const v2f_g*)(Jb + (size_t)row * KDIM + 4);
            ev  = eb[row];
            wv  = wb[row / DDIM];
        }

        // ---- 8 WMMAs over the 32 staged rows, immediate LDS offsets ----
        const float* jc = jcol + buf * LSLICE;
        const float* wc = wcol + buf * LSLICE;
        #pragma unroll
        for (int k = 0; k < 8; k += 2) {
            {
                const int off = k * 4 * LROW;                 // rows k*4 ..
                const float v0 = jc[off], v1 = jc[off + LROW];
                const float w0 = wc[off], w1 = wc[off + LROW];
                v2f a, bf;
                a.x  = w0 * v0;  a.y  = w1 * v1;
                bf.x = v0;       bf.y = v1;
                c0 = __builtin_amdgcn_wmma_f32_16x16x4_f32(
                         false, a, false, bf, (short)0, c0, false, false);
            }
            {
                const int off = (k + 1) * 4 * LROW;           // rows (k+1)*4 ..
                const float v0 = jc[off], v1 = jc[off + LROW];
                const float w0 = wc[off], w1 = wc[off + LROW];
                v2f a, bf;
                a.x  = w0 * v0;  a.y  = w1 * v1;
                bf.x = v0;       bf.y = v1;
                c1 = __builtin_amdgcn_wmma_f32_16x16x4_f32(
                         false, a, false, bf, (short)0, c1, false, false);
            }
        }
    }

    c0 += c1;  // merge the two accumulator chains

    // C/D layout: lane L, VGPR v holds D[v + 8*(L/16)][L%16].
    // Useful block: rows m=0..5 (v=0..5, lanes 0-15), cols n=0..6 (lane 0..6).
    if (lane < KDIM + 1) {
        float* accb = acc + b * ACC_STRIDE;
        #pragma unroll
        for (int v = 0; v < KDIM; ++v) {
            atomicAdd(&accb[v * 8 + lane], c0[v]);
        }
    }
}

// ---------------------------------------------------------------------------
// Kernel 3: per-batch LM damping + 6x6 Cholesky solve + output x + delta.
// One thread per batch: 1024 threads total, negligible cost.
// ---------------------------------------------------------------------------
__global__ void gn_solve(const float* __restrict__ x,
                         const float* __restrict__ acc,
                         const float* __restrict__ renta,
                         float* __restrict__ out) {
    const int b = blockIdx.x * blockDim.x + threadIdx.x;
    if (b >= BS) return;

    const float* ab = acc + b * ACC_STRIDE;
    float A[KDIM][KDIM];
    float g[KDIM];
    #pragma unroll
    for (int i = 0; i < KDIM; ++i) {
        #pragma unroll
        for (int j = 0; j < KDIM; ++j) A[i][j] = ab[i * 8 + j];
        g[i] = ab[i * 8 + KDIM];
    }

    // Levenberg-Marquardt damping: diag += clip(diag * renta, min=DIAG_MIN)
    const float rn = renta[0];
    #pragma unroll
    for (int k = 0; k < KDIM; ++k) {
        float d = A[k][k] * rn;
        d = fmaxf(d, DIAG_MIN);
        A[k][k] += d;
    }

    // Cholesky factorization A = L L^T (A is SPD after damping)
    float L[KDIM][KDIM];
    #pragma unroll
    for (int i = 0; i < KDIM; ++i) {
        #pragma unroll
        for (int j = 0; j <= i; ++j) {
            float s = A[i][j];
            #pragma unroll
            for (int p = 0; p < KDIM; ++p)
                if (p < j) s -= L[i][p] * L[j][p];
            if (i == j) L[i][i] = sqrtf(s);
            else        L[i][j] = s / L[j][j];
        }
    }

    // Forward substitution: L y = g
    float y[KDIM];
    #pragma unroll
    for (int i = 0; i < KDIM; ++i) {
        float s = g[i];
        #pragma unroll
        for (int p = 0; p < KDIM; ++p)
            if (p < i) s -= L[i][p] * y[p];
        y[i] = s / L[i][i];
    }

    // Back substitution: L^T delta = y
    float dlt[KDIM];
    #pragma unroll
    for (int i = KDIM - 1; i >= 0; --i) {
        float s = y[i];
        #pragma unroll
        for (int p = 0; p < KDIM; ++p)
            if (p > i) s -= L[p][i] * dlt[p];
        dlt[i] = s / L[i][i];
    }

    #pragma unroll
    for (int k = 0; k < KDIM; ++k)
        out[b * KDIM + k] = x[b * KDIM + k] + dlt[k];
}

// ---------------------------------------------------------------------------
extern "C" void kernel_launch(void* const* d_in, const int* in_sizes, int n_in,
                              void* d_out, int out_size, void* d_ws, size_t ws_size,
                              hipStream_t stream) {
    const float* x      = (const float*)d_in[0];   // [BS, 6]
    const float* e      = (const float*)d_in[1];   // [BS, N, 3]
    const float* J      = (const float*)d_in[2];   // [BS, N, 3, 6]
    const float* weight = (const float*)d_in[3];   // [BS, N, 1]
    const float* renta  = (const float*)d_in[4];   // [1]
    float*       out    = (float*)d_out;           // [BS, 6]
    float*       acc    = (float*)d_ws;            // BS * 48 floats = 192 KB

    gn_zero_acc<<<(BS * ACC_STRIDE + 255) / 256, 256, 0, stream>>>(acc);
    gn_accumulate<<<BS * BLOCKS_PER_BATCH, THREADS, 0, stream>>>(e, J, weight, acc);
    gn_solve<<<(BS + 255) / 256, 256, 0, stream>>>(x, acc, renta, out);
}